// PET_layer_58892591563020
// MI455X (gfx1250) — compile-verified
//
#include <hip/hip_runtime.h>

typedef float v2f __attribute__((ext_vector_type(2)));
typedef float v8f __attribute__((ext_vector_type(8)));

#define BB 128
#define SS 512
#define DD 1024
#define START_ID 30522
#define END_ID   30523

// ---------------------------------------------------------------------------
// Kernel 1: find span markers, gather-sum span rows -> euph [B, D]
// One block per batch row; 256 threads; each thread owns 4 contiguous floats
// (one float4) of the D dimension and walks the <=31-row span.
// ---------------------------------------------------------------------------
__global__ void span_sum_kernel(const float* __restrict__ inputs,
                                const int*   __restrict__ ids,
                                float*       __restrict__ euph) {
  __shared__ int s_start, s_end;
  const int b = blockIdx.x;
  const int t = threadIdx.x;

  if (t == 0) { s_start = 0; s_end = 0; }
  __syncthreads();

  const int* row = ids + (size_t)b * SS;
  for (int s = t; s < SS; s += 256) {
    const int id = row[s];
    if (id == START_ID) s_start = s;   // unique per row (single writer)
    else if (id == END_ID) s_end = s;  // unique per row (single writer)
  }
  __syncthreads();

  const int lo = s_start + 1;
  const int hi = s_end;                 // strictly-between span: [lo, hi)

  float4 acc = make_float4(0.f, 0.f, 0.f, 0.f);
  const float* base = inputs + (size_t)b * SS * DD + t * 4;
  for (int s = lo; s < hi; ++s) {
    const float4 v = *(const float4*)(base + (size_t)s * DD);
    acc.x += v.x; acc.y += v.y; acc.z += v.z; acc.w += v.w;
  }
  *(float4*)(euph + (size_t)b * DD + t * 4) = acc;
}

// ---------------------------------------------------------------------------
// Kernel 2: h = euph @ W1^T + b1 via V_WMMA_F32_16X16X4_F32.
// 8 waves / block, 64 blocks -> 512 waves, one 16x16 output tile each
// (M-tiles = 128/16 = 8, N-tiles = 1024/16 = 64). K loop steps of 4.
// A-fragment (16x4):  lane l -> row m0+(l&15), K = k0 + 2*(l>>4) + {0,1}
// B-fragment (4x16):  lane l -> col n0+(l&15), K = k0 + 2*(l>>4) + {0,1}
//                     B[k][n] = W1[n][k]  => contiguous float2 from W1 row n.
// C/D (16x16 f32):    VGPR v, lane l -> row m0 + v + 8*(l>>4), col n0+(l&15)
// No divergence anywhere: EXEC is all-1s as WMMA requires.
// ---------------------------------------------------------------------------
__global__ void gemm_wmma_kernel(const float* __restrict__ A,    // euph [B, D]
                                 const float* __restrict__ W1,   // [D, D]
                                 const float* __restrict__ b1,   // [D]
                                 float*       __restrict__ H) {  // [B, D]
  const int wave = threadIdx.x >> 5;           // 0..7
  const int lane = threadIdx.x & 31;
  const int tile = blockIdx.x * 8 + wave;      // 0..511
  const int m0 = (tile >> 6) * 16;             // 8 M-tiles
  const int n0 = (tile & 63) * 16;             // 64 N-tiles
  const int half = lane >> 4;                  // 0 -> K{0,1}, 1 -> K{2,3}
  const int l15  = lane & 15;

  const float* arow = A  + (size_t)(m0 + l15) * DD + 2 * half;
  const float* brow = W1 + (size_t)(n0 + l15) * DD + 2 * half;

  v8f c = {};
#pragma unroll 4
  for (int k = 0; k < DD; k += 4) {
    const v2f a  = *(const v2f*)(arow + k);
    const v2f bv = *(const v2f*)(brow + k);
    // 8 args: (neg_a, A, neg_b, B, c_mod, C, reuse_a, reuse_b)
    c = __builtin_amdgcn_wmma_f32_16x16x4_f32(false, a, false, bv,
                                              (short)0, c, false, false);
  }

  const float bias = b1[n0 + l15];
  float* out = H + (size_t)(m0 + 8 * half) * DD + n0 + l15;
#pragma unroll
  for (int v = 0; v < 8; ++v) {
    out[(size_t)v * DD] = c[v] + bias;
  }
}

// ---------------------------------------------------------------------------
// Kernel 3: out[b, {0,1}] = h[b,:] . W2[{0,1},:] + b2  (tiny head GEMV)
// One block per batch row; dual accumulate + shared reduction.
// ---------------------------------------------------------------------------
__global__ void head_kernel(const float* __restrict__ H,    // [B, D]
                            const float* __restrict__ W2,   // [2, D]
                            const float* __restrict__ b2,   // [2]
                            float*       __restrict__ out) {// [B, 2]
  __shared__ float r0[256];
  __shared__ float r1[256];
  const int b = blockIdx.x;
  const int t = threadIdx.x;

  const float* h = H + (size_t)b * DD;
  float a0 = 0.f, a1 = 0.f;
  for (int d = t; d < DD; d += 256) {
    const float hv = h[d];
    a0 += hv * W2[d];
    a1 += hv * W2[DD + d];
  }
  r0[t] = a0; r1[t] = a1;
  __syncthreads();
  for (int s = 128; s > 0; s >>= 1) {
    if (t < s) { r0[t] += r0[t + s]; r1[t] += r1[t + s]; }
    __syncthreads();
  }
  if (t == 0) {
    out[b * 2 + 0] = r0[0] + b2[0];
    out[b * 2 + 1] = r1[0] + b2[1];
  }
}

// ---------------------------------------------------------------------------
// Launcher. Input order (setup_inputs dict): inputs, input_ids, W1, b1, W2, b2
// Workspace: euph [128*1024 f32] at offset 0, h [128*1024 f32] after it.
// ---------------------------------------------------------------------------
extern "C" void kernel_launch(void* const* d_in, const int* in_sizes, int n_in,
                              void* d_out, int out_size, void* d_ws, size_t ws_size,
                              hipStream_t stream) {
  (void)in_sizes; (void)n_in; (void)out_size; (void)ws_size;

  const float* inputs = (const float*)d_in[0];
  const int*   ids    = (const int*)  d_in[1];
  const float* W1     = (const float*)d_in[2];
  const float* b1     = (const float*)d_in[3];
  const float* W2     = (const float*)d_in[4];
  const float* b2     = (const float*)d_in[5];
  float*       out    = (float*)d_out;

  float* euph = (float*)d_ws;                       // 128*1024 floats
  float* h    = euph + (size_t)BB * DD;             // 128*1024 floats

  span_sum_kernel <<<BB, 256, 0, stream>>>(inputs, ids, euph);
  gemm_wmma_kernel<<<64, 256, 0, stream>>>(euph, W1, b1, h);
  head_kernel     <<<BB, 256, 0, stream>>>(h, W2, b2, out);
}